// L_smooth_ill_20126216749853
// MI455X (gfx1250) — compile-verified
//
#include <hip/hip_runtime.h>
#include <math.h>

#define BATCH 16
#define Hh 512
#define Ww 512
#define HW (Hh * Ww)
#define NTOT (BATCH * HW)
#define EPSV 1e-4f
#define RED_BLOCKS 4096

typedef __attribute__((ext_vector_type(2))) float v2f;
typedef __attribute__((ext_vector_type(8))) float v8f;

// ---------- grayscale for both inputs ----------
__global__ void k_gray(const float* __restrict__ Rin, const float* __restrict__ Lin,
                       float* __restrict__ Rg, float* __restrict__ Lg) {
    int i = blockIdx.x * blockDim.x + threadIdx.x;
    if (i >= NTOT) return;
    int b = i / HW;
    int rem = i - b * HW;
    long base = (long)b * 3 * HW + rem;
    Rg[i] = 0.299f * Rin[base] + 0.587f * Rin[base + HW] + 0.114f * Rin[base + 2 * HW];
    Lg[i] = 0.299f * Lin[base] + 0.587f * Lin[base + HW] + 0.114f * Lin[base + 2 * HW];
}

// ---------- gradient (dir 0 = x/last dim, dir 1 = y/h dim); also emit |g| ----------
__global__ void k_grad(const float* __restrict__ in, float* __restrict__ g,
                       float* __restrict__ ga, int dir) {
    int i = blockIdx.x * blockDim.x + threadIdx.x;
    if (i >= NTOT) return;
    int w = i % Ww;
    int h = (i / Ww) % Hh;
    float nb;
    if (dir == 0) nb = (w + 1 < Ww) ? in[i + 1] : 0.0f;
    else          nb = (h + 1 < Hh) ? in[i + Ww] : 0.0f;
    float gv = in[i] - nb;
    g[i] = gv;
    ga[i] = fabsf(gv);
}

// ---------- row-direction sliding max & min of one array ----------
__global__ void k_row_minmax(const float* __restrict__ in, float* __restrict__ omax,
                             float* __restrict__ omin, int r) {
    int i = blockIdx.x * blockDim.x + threadIdx.x;
    if (i >= NTOT) return;
    int w = i % Ww;
    long rowbase = (long)(i / Ww) * Ww;
    int lo = w - r; if (lo < 0) lo = 0;
    int hi = w + r; if (hi > Ww - 1) hi = Ww - 1;
    float mx = -INFINITY, mn = INFINITY;
    for (int t = lo; t <= hi; ++t) {
        float v = in[rowbase + t];
        mx = fmaxf(mx, v);
        mn = fminf(mn, v);
    }
    omax[i] = mx;
    omin[i] = mn;
}

// ---------- col-direction sliding max of A and min of B ----------
__global__ void k_col_minmax(const float* __restrict__ inA, const float* __restrict__ inB,
                             float* __restrict__ omax, float* __restrict__ omin, int r) {
    int i = blockIdx.x * blockDim.x + threadIdx.x;
    if (i >= NTOT) return;
    int w = i % Ww;
    int h = (i / Ww) % Hh;
    int b = i / HW;
    int lo = h - r; if (lo < 0) lo = 0;
    int hi = h + r; if (hi > Hh - 1) hi = Hh - 1;
    float mx = -INFINITY, mn = INFINITY;
    long base = (long)b * HW + w;
    for (int t = lo; t <= hi; ++t) {
        mx = fmaxf(mx, inA[base + (long)t * Ww]);
        mn = fminf(mn, inB[base + (long)t * Ww]);
    }
    omax[i] = mx;
    omin[i] = mn;
}

// ---------- row-direction windowed sum / cnt_w ----------
__global__ void k_row_sum(const float* __restrict__ in, float* __restrict__ out, int r) {
    int i = blockIdx.x * blockDim.x + threadIdx.x;
    if (i >= NTOT) return;
    int w = i % Ww;
    long rowbase = (long)(i / Ww) * Ww;
    int lo = w - r; if (lo < 0) lo = 0;
    int hi = w + r; if (hi > Ww - 1) hi = Ww - 1;
    float s = 0.0f;
    for (int t = lo; t <= hi; ++t) s += in[rowbase + t];
    out[i] = s / (float)(hi - lo + 1);
}

// ---------- fused: col-sum17/cnt_h of Smax,Smin -> gmax1,gmin1; gn1 ----------
__global__ void k_gn1(const float* __restrict__ Smax, const float* __restrict__ Smin,
                      const float* __restrict__ ga, float* __restrict__ gn1) {
    const int r = 8;
    int i = blockIdx.x * blockDim.x + threadIdx.x;
    if (i >= NTOT) return;
    int w = i % Ww;
    int h = (i / Ww) % Hh;
    int b = i / HW;
    int lo = h - r; if (lo < 0) lo = 0;
    int hi = h + r; if (hi > Hh - 1) hi = Hh - 1;
    float smax = 0.0f, smin = 0.0f;
    long base = (long)b * HW + w;
    for (int t = lo; t <= hi; ++t) {
        smax += Smax[base + (long)t * Ww];
        smin += Smin[base + (long)t * Ww];
    }
    float cnt = (float)(hi - lo + 1);
    float gmax1 = smax / cnt;
    float gmin1 = smin / cnt;
    gn1[i] = (ga[i] - gmin1) / (fabsf(gmax1 - gmin1) + EPSV);
}

// ---------- fused: col-sum5/cnt_h + abs -> go ----------
__global__ void k_go(const float* __restrict__ in, float* __restrict__ out) {
    const int r = 2;
    int i = blockIdx.x * blockDim.x + threadIdx.x;
    if (i >= NTOT) return;
    int w = i % Ww;
    int h = (i / Ww) % Hh;
    int b = i / HW;
    int lo = h - r; if (lo < 0) lo = 0;
    int hi = h + r; if (hi > Hh - 1) hi = Hh - 1;
    float s = 0.0f;
    long base = (long)b * HW + w;
    for (int t = lo; t <= hi; ++t) s += in[base + (long)t * Ww];
    out[i] = fabsf(s / (float)(hi - lo + 1));
}

// ---------- fused: col-sum7/cnt_h of Smax,Smin -> pmax,pmin; gn; field ----------
__global__ void k_field(const float* __restrict__ Smax, const float* __restrict__ Smin,
                        const float* __restrict__ go, const float* __restrict__ gn1,
                        float* __restrict__ field) {
    const int r = 3;
    int i = blockIdx.x * blockDim.x + threadIdx.x;
    if (i >= NTOT) return;
    int w = i % Ww;
    int h = (i / Ww) % Hh;
    int b = i / HW;
    int lo = h - r; if (lo < 0) lo = 0;
    int hi = h + r; if (hi > Hh - 1) hi = Hh - 1;
    float smax = 0.0f, smin = 0.0f;
    long base = (long)b * HW + w;
    for (int t = lo; t <= hi; ++t) {
        smax += Smax[base + (long)t * Ww];
        smin += Smin[base + (long)t * Ww];
    }
    float cnt = (float)(hi - lo + 1);
    float pmax = smax / cnt;
    float pmin = smin / cnt;
    float gn = (go[i] - pmin) / (fabsf(pmax - pmin) + EPSV);
    field[i] = (gn + 0.01f) * gn1[i];
}

// ---------- reduction: sum of Rf*exp(-10*Rf)*exp(-10*Lf), WMMA block-reduce ----------
__global__ void k_reduce(const float* __restrict__ Rf, const float* __restrict__ Lf,
                         float* __restrict__ partials) {
    __shared__ float lds[256];
    int tid = threadIdx.x;
    float s = 0.0f;
    int stride = gridDim.x * blockDim.x;           // 4096*256 = 1,048,576
    for (int i = blockIdx.x * blockDim.x + tid; i < NTOT; i += stride) {
        float a = Rf[i];
        float c = Lf[i];
        s += a * expf(-10.0f * (a + c));
    }
    lds[tid] = s;
    __syncthreads();
    if (tid < 32) {
        // Sum 256 values with 4 accumulating V_WMMA_F32_16X16X4_F32 (B = ones).
        // Any bijective lane->A-element packing works: D[m][n] = sum_k A[m][k].
        v2f ones2 = {1.0f, 1.0f};
        v8f acc = {};
        for (int t = 0; t < 4; ++t) {
            v2f a;
            a.x = lds[t * 64 + tid];
            a.y = lds[t * 64 + 32 + tid];
            acc = __builtin_amdgcn_wmma_f32_16x16x4_f32(
                false, a, false, ones2, (short)0, acc, false, false);
        }
        float sd = acc[0] + acc[1] + acc[2] + acc[3] + acc[4] + acc[5] + acc[6] + acc[7];
        // columns are identical; rows 0..7 live on lanes 0..15, rows 8..15 on lanes 16..31
        float tot = __shfl(sd, 0, 32) + __shfl(sd, 16, 32);
        if (tid == 0) partials[blockIdx.x] = tot;
    }
}

// ---------- deterministic finalize: out = (dir==0 ? 0 : out) + sum(partials)/N ----------
__global__ void k_finalize(const float* __restrict__ partials, float* __restrict__ out, int dir) {
    __shared__ float lds[256];
    int tid = threadIdx.x;
    float s = 0.0f;
    for (int i = tid; i < RED_BLOCKS; i += 256) s += partials[i];
    lds[tid] = s;
    __syncthreads();
    for (int off = 128; off > 0; off >>= 1) {
        if (tid < off) lds[tid] += lds[tid + off];
        __syncthreads();
    }
    if (tid == 0) {
        float m = lds[0] / (float)NTOT;
        if (dir == 0) out[0] = m;
        else          out[0] += m;
    }
}

extern "C" void kernel_launch(void* const* d_in, const int* in_sizes, int n_in,
                              void* d_out, int out_size, void* d_ws, size_t ws_size,
                              hipStream_t stream) {
    const float* R_low = (const float*)d_in[0];
    const float* low   = (const float*)d_in[1];
    float* out = (float*)d_out;

    // workspace layout: 10 planes of NTOT floats + RED_BLOCKS partials (~160 MiB)
    float* wsf = (float*)d_ws;
    const size_t P = (size_t)NTOT;
    float* Rg = wsf + 0 * P;
    float* Lg = wsf + 1 * P;
    float* F0 = wsf + 2 * P;   // field from Rg for current direction
    float* F1 = wsf + 3 * P;   // field from Lg for current direction
    float* B0 = wsf + 4 * P;
    float* B1 = wsf + 5 * P;
    float* B2 = wsf + 6 * P;
    float* B3 = wsf + 7 * P;
    float* B4 = wsf + 8 * P;
    float* B5 = wsf + 9 * P;
    float* partials = wsf + 10 * P;

    const int TB = 256;
    const int GB = (NTOT + TB - 1) / TB;   // 16384

    k_gray<<<GB, TB, 0, stream>>>(R_low, low, Rg, Lg);

    for (int dir = 0; dir < 2; ++dir) {
        for (int img = 0; img < 2; ++img) {
            const float* src = (img == 0) ? Rg : Lg;
            float* field     = (img == 0) ? F0 : F1;

            // g -> B0, |g| -> B1
            k_grad<<<GB, TB, 0, stream>>>(src, B0, B1, dir);

            // ---- gn1 branch ----
            k_row_minmax<<<GB, TB, 0, stream>>>(B1, B2, B3, 4);         // row max/min 9
            k_col_minmax<<<GB, TB, 0, stream>>>(B2, B3, B4, B5, 4);     // pmax1->B4, pmin1->B5
            k_row_sum<<<GB, TB, 0, stream>>>(B4, B2, 8);                // row sum17/cnt_w
            k_row_sum<<<GB, TB, 0, stream>>>(B5, B3, 8);
            k_gn1<<<GB, TB, 0, stream>>>(B2, B3, B1, B4);               // gn1 -> B4

            // ---- gn branch ----
            k_row_sum<<<GB, TB, 0, stream>>>(B0, B2, 2);                // row sum5/cnt_w of g
            k_go<<<GB, TB, 0, stream>>>(B2, B3);                        // go -> B3
            k_row_minmax<<<GB, TB, 0, stream>>>(B3, B0, B2, 3);         // row max/min 7 of go
            k_col_minmax<<<GB, TB, 0, stream>>>(B0, B2, B5, B1, 3);     // maxpool7->B5, minpool7->B1
            k_row_sum<<<GB, TB, 0, stream>>>(B5, B0, 3);                // row sum7/cnt_w
            k_row_sum<<<GB, TB, 0, stream>>>(B1, B2, 3);
            k_field<<<GB, TB, 0, stream>>>(B0, B2, B3, B4, field);      // (gn+0.01)*gn1
        }
        k_reduce<<<RED_BLOCKS, TB, 0, stream>>>(F0, F1, partials);
        k_finalize<<<1, TB, 0, stream>>>(partials, out, dir);
    }
}